// YoloLossLayer_71614284693898
// MI455X (gfx1250) — compile-verified
//
#include <hip/hip_runtime.h>
#include <hip/hip_bf16.h>
#include <stdint.h>

// YOLO-v3 loss on MI455X (gfx1250). Memory-bound streaming reduction:
// ~231 MB read once -> ~10 us floor at 23.3 TB/s (~180 GB/s per WGP, needing
// ~128 KB in flight per WGP). All global reads go through the CDNA5 async
// global->LDS engine (ASYNCcnt: up to 63 outstanding instrs/wave, 512 B each,
// no VGPR dest pressure); compute is DS+VALU only, with hardware
// v_exp_f32/v_log_f32 for the 240 BCE softplus terms per thread so VALU time
// stays under the streaming time. No GEMM structure => WMMA does not apply.
// Two 130 KB-LDS blocks per WGP overlap stage/compute.

#define CPB 64            // cells per block == blockDim.x (2 waves)
#define IMG_SIZE 416.0f

__global__ void yolo_ws_init(float* __restrict__ ws) {
  int t = threadIdx.x;
  if (t < 18) ws[t] = 0.0f;
}

__device__ __forceinline__ float bce_term(float x, float t) {
  // max(x,0) - x*t + log1p(exp(-|x|)); arg of exp is <= 0 so no overflow,
  // and 1+e loses only terms < 2^-24 (abs err < 6e-8 on O(1) loss values).
  return fmaxf(x, 0.0f) - x * t + __logf(1.0f + __expf(-fabsf(x)));
}

__device__ __forceinline__ float sigmoidf_fast(float x) {
  return 1.0f / (1.0f + __expf(-x));   // exp(-x)->inf gives exact 0: fine
}

__device__ __forceinline__ void async_b128(const float* src, const float* ldsDst) {
  unsigned lds_addr = (unsigned)(uintptr_t)ldsDst;
  unsigned long long ga = (unsigned long long)(uintptr_t)src;
  asm volatile("global_load_async_to_lds_b128 %0, %1, off"
               :: "v"(lds_addr), "v"(ga) : "memory");
}
__device__ __forceinline__ void async_b32(const float* src, const float* ldsDst) {
  unsigned lds_addr = (unsigned)(uintptr_t)ldsDst;
  unsigned long long ga = (unsigned long long)(uintptr_t)src;
  asm volatile("global_load_async_to_lds_b32 %0, %1, off"
               :: "v"(lds_addr), "v"(ga) : "memory");
}

__launch_bounds__(CPB)
__global__ void yolo_scale_kernel(const float* __restrict__ fm,
                                  const float* __restrict__ yt,
                                  const float* __restrict__ anc,   // 3 (w,h) pairs
                                  float* __restrict__ ws6,         // 6 accumulators
                                  int G, int B) {
  __shared__ __align__(16) float fmtile[255 * CPB];  // [ch][cell]  (65,280 B)
  __shared__ __align__(16) float ytile[255 * CPB];   // [cell][ch]  (65,280 B)
  const int GG = G * G;
  const int b = blockIdx.y;
  const int tileStart = blockIdx.x * CPB;
  const int tid = threadIdx.x;
  const int nChunks = (255 * CPB) / 4;               // 4080 x 16 B per tensor

  // ---- async stage y_true rows: contiguous 64*255 floats, 16 B per lane ----
  {
    const long ybase = ((long)b * GG + tileStart) * 255;  // float offset
    const long bufEnd = (long)B * GG * 255;               // total y floats
    for (int q = tid; q < nChunks; q += CPB) {
      const int rel = q * 4;
      const long src = ybase + rel;
      if (src + 4 <= bufEnd) {
        async_b128(yt + src, &ytile[rel]);
      } else if (src < bufEnd) {            // last chunk of whole buffer
        for (int j = 0; j < 4; ++j) {
          long sj = src + j; if (sj >= bufEnd) sj = bufEnd - 1;
          async_b32(yt + sj, &ytile[rel + j]);
        }
      }                                      // fully past end: slots unused
    }
  }

  // ---- async stage fm: per channel, 64 consecutive cells of this plane ----
  {
    const float* fmb = fm + (size_t)b * 255 * GG;
    for (int q = tid; q < nChunks; q += CPB) {
      const int ch = q >> 4;
      const int n0 = tileStart + ((q & 15) << 2);
      if (n0 + 4 <= GG) {
        async_b128(fmb + (size_t)ch * GG + n0, &fmtile[ch * CPB + (n0 - tileStart)]);
      } else {
        for (int j = 0; j < 4; ++j) {        // plane-boundary chunk, exact
          const int nj = n0 + j;
          if (nj < GG)
            async_b32(fmb + (size_t)ch * GG + nj, &fmtile[ch * CPB + (nj - tileStart)]);
        }
      }
    }
  }
  asm volatile("s_wait_asynccnt 0" ::: "memory");
  __syncthreads();

  const int n = tileStart + tid;
  float s1 = 0.f, s2 = 0.f, sbc = 0.f, s3 = 0.f, sbp = 0.f, s4 = 0.f;

  if (n < GG) {
    const int gi = n / G;          // row    -> y offset
    const int gj = n - gi * G;     // column -> x offset
    const float inv_ratio = (float)G / IMG_SIZE;
    const float* __restrict__ yr = &ytile[tid * 255];  // banks (k - tid) % 64
#define FM(ch) fmtile[(ch) * CPB + tid]                /* bank == tid         */

    // ignore-mask 'sel': wh of the LAST anchor slot with nonzero wh (else 0)
    float selw = 0.f, selh = 0.f;
#pragma unroll
    for (int a = 0; a < 3; ++a) {
      float w = yr[a * 85 + 2], h = yr[a * 85 + 3];
      if (w != 0.f || h != 0.f) { selw = w; selh = h; }
    }

#pragma unroll
    for (int a = 0; a < 3; ++a) {
      const int fc = a * 85;
      const float f0 = FM(fc + 0);
      const float f1 = FM(fc + 1);
      const float f2 = FM(fc + 2);
      const float f3 = FM(fc + 3);
      const float f4 = FM(fc + 4);

      const float tx = yr[fc + 0], ty = yr[fc + 1];
      const float tw = yr[fc + 2], th = yr[fc + 3];
      const float obj = yr[fc + 4];
      const float aw = anc[2 * a], ah = anc[2 * a + 1];

      // xy loss: pred_xy == sigmoid(fm[0:2]) (grid offsets cancel)
      const float px = sigmoidf_fast(f0);
      const float py = sigmoidf_fast(f1);
      const float dx = (tx * inv_ratio - (float)gj) - px;
      const float dy = (ty * inv_ratio - (float)gi) - py;
      const float bls = 2.0f - (tw * (1.0f / IMG_SIZE)) * (th * (1.0f / IMG_SIZE));
      s1 += (dx * dx + dy * dy) * obj * bls;

      // wh loss (log-clipped) - low count, keep precise libm path
      const float t_tw = logf(fminf(fmaxf(tw / aw, 1e-9f), 1e9f));
      const float t_th = logf(fminf(fmaxf(th / ah, 1e-9f), 1e9f));
      const float p_tw = logf(fminf(fmaxf(expf(f2), 1e-9f), 1e9f));
      const float p_th = logf(fminf(fmaxf(expf(f3), 1e-9f), 1e9f));
      const float dw = t_tw - p_tw, dh = t_th - p_th;
      s2 += (dw * dw + dh * dh) * obj * bls;

      // conf: mean-BCE numerator + focal*conf_mask sum
      sbc += bce_term(f4, obj);
      const float sg = sigmoidf_fast(f4);
      float fo = obj - sg; fo = fo * fo;              // |.|^2 == square
      const float inter = fminf(selw, aw) * fminf(selh, ah);
      const float iou = inter / (selw * selh + aw * ah - inter + 1e-10f);
      const float ign = (iou > 0.5f) ? 0.0f : 1.0f;
      s3 += (obj + (1.0f - obj) * ign) * fo;
      s4 += obj;

      // class BCE (mean numerator) with label smoothing - the hot 240 terms
#pragma unroll 8
      for (int c = 0; c < 80; ++c) {
        const float p = FM(fc + 5 + c);
        const float t = 0.99f * yr[fc + 5 + c] + (0.01f / 80.0f);
        sbp += bce_term(p, t);
      }
    }
#undef FM
  }

  // ---- wave32 reduction, then one f32 atomic per wave per accumulator ----
#pragma unroll
  for (int off = 16; off > 0; off >>= 1) {
    s1  += __shfl_down(s1,  off, 32);
    s2  += __shfl_down(s2,  off, 32);
    sbc += __shfl_down(sbc, off, 32);
    s3  += __shfl_down(s3,  off, 32);
    sbp += __shfl_down(sbp, off, 32);
    s4  += __shfl_down(s4,  off, 32);
  }
  if ((tid & 31) == 0) {
    atomicAdd(ws6 + 0, s1);
    atomicAdd(ws6 + 1, s2);
    atomicAdd(ws6 + 2, sbc);
    atomicAdd(ws6 + 3, s3);
    atomicAdd(ws6 + 4, sbp);
    atomicAdd(ws6 + 5, s4);
  }
}

__global__ void yolo_finalize(const float* __restrict__ ws,
                              float* __restrict__ out, int B) {
  if (threadIdx.x == 0 && blockIdx.x == 0) {
    const int Gs[3] = {13, 26, 52};
    float xy = 0.f, wh = 0.f, conf = 0.f, prob = 0.f;
    const float invB = 1.0f / (float)B;
#pragma unroll
    for (int s = 0; s < 3; ++s) {
      const float S1 = ws[s * 6 + 0];
      const float S2 = ws[s * 6 + 1];
      const float BC = ws[s * 6 + 2];
      const float S3 = ws[s * 6 + 3];
      const float BP = ws[s * 6 + 4];
      const float S4 = ws[s * 6 + 5];
      const float cnt = (float)B * (float)(Gs[s] * Gs[s]) * 3.0f;
      xy   += S1 * invB;
      wh   += S2 * invB;
      conf += (BC / cnt) * S3 * invB;
      prob += (BP / (cnt * 80.0f)) * S4 * invB;
    }
    out[0] = xy + wh + conf + prob;
    out[1] = xy;
    out[2] = wh;
    out[3] = conf;
    out[4] = prob;
  }
}

extern "C" void kernel_launch(void* const* d_in, const int* in_sizes, int n_in,
                              void* d_out, int out_size, void* d_ws, size_t ws_size,
                              hipStream_t stream) {
  (void)n_in; (void)out_size; (void)ws_size;
  const float* fm[3] = {(const float*)d_in[0], (const float*)d_in[1], (const float*)d_in[2]};
  const float* yy[3] = {(const float*)d_in[3], (const float*)d_in[4], (const float*)d_in[5]};
  const float* anchors = (const float*)d_in[6];
  float* ws = (float*)d_ws;
  float* out = (float*)d_out;

  const int B = in_sizes[0] / (255 * 13 * 13);
  const int Gs[3]   = {13, 26, 52};
  const int aoff[3] = {12, 6, 0};   // groups: [anchors[6:9], anchors[3:6], anchors[0:3]]

  yolo_ws_init<<<1, 32, 0, stream>>>(ws);

  for (int s = 0; s < 3; ++s) {
    const int GG = Gs[s] * Gs[s];
    const int tiles = (GG + CPB - 1) / CPB;
    dim3 grid(tiles, B);
    yolo_scale_kernel<<<grid, CPB, 0, stream>>>(
        fm[s], yy[s], anchors + aoff[s], ws + s * 6, Gs[s], B);
  }

  yolo_finalize<<<1, 1, 0, stream>>>(ws, out, B);
}